// MultiLevelEmbedding_24902220382934
// MI455X (gfx1250) — compile-verified
//
#include <hip/hip_runtime.h>
#include <math.h>

// MultiLevelEmbedding: gather(emb0)+gather(emb1) || gather(pos_table) -> LayerNorm.
// Memory-bound: ~768MB traffic -> ~33us at 23.3 TB/s. emb0_w (103MB) fits in the
// 192MB L2, so outputs are written with non-temporal hints to avoid evicting it.

typedef float v4f __attribute__((ext_vector_type(4)));
typedef int   v4i_gcc __attribute__((vector_size(16)));   // matches builtin param type
typedef __attribute__((address_space(1))) v4i_gcc* gptr_v4i;
typedef __attribute__((address_space(3))) v4i_gcc* lptr_v4i;

#define T_TOK 65536
#define D_C   512          // content width == pos width
#define D_E   1024         // concat width
#define WPB   8            // waves per 256-thread block (wave32)
#define NBLK  1024         // 8192 waves total; each wave does 8 rows (prefetch pipeline)
#define EPS   1e-3f

#if __has_builtin(__builtin_amdgcn_global_load_async_to_lds_b128)
#define HAVE_ASYNC_LDS 1
#else
#define HAVE_ASYNC_LDS 0
#endif

__global__ __launch_bounds__(256) void mle_kernel(
    const int* __restrict__ x0, const int* __restrict__ x1,
    const int* __restrict__ pos_ids,
    const float* __restrict__ emb0, const float* __restrict__ emb1,
    const float* __restrict__ pos_table,
    const float* __restrict__ a2, const float* __restrict__ b2,
    float* __restrict__ ann, float* __restrict__ timing_out)
{
    // ---- stage a2 || b2 into LDS once per block (gfx1250 async DMA path) ----
    __shared__ __align__(16) float s_ab[2 * D_E];
    const int tid = threadIdx.x;           // 0..255, each thread moves 16B per table
#if HAVE_ASYNC_LDS
    __builtin_amdgcn_global_load_async_to_lds_b128(
        (gptr_v4i)(a2 + tid * 4),
        (lptr_v4i)(&s_ab[tid * 4]), 0, 0);
    __builtin_amdgcn_global_load_async_to_lds_b128(
        (gptr_v4i)(b2 + tid * 4),
        (lptr_v4i)(&s_ab[D_E + tid * 4]), 0, 0);
#if __has_builtin(__builtin_amdgcn_s_wait_asynccnt)
    __builtin_amdgcn_s_wait_asynccnt(0);
#else
    asm volatile("s_wait_asynccnt 0" ::: "memory");
#endif
#else
    *(v4f*)(&s_ab[tid * 4])       = *(const v4f*)(a2 + tid * 4);
    *(v4f*)(&s_ab[D_E + tid * 4]) = *(const v4f*)(b2 + tid * 4);
#endif
    __syncthreads();

    const int wave = tid >> 5;
    const int lane = tid & 31;

    // Per-lane gamma/beta slices live in VGPRs for the whole kernel (ds_load_b128 x16).
    v4f a2c[4], a2t[4], b2c[4], b2t[4];
#pragma unroll
    for (int j = 0; j < 4; ++j) {
        a2c[j] = *(const v4f*)(&s_ab[lane * 16 + j * 4]);
        a2t[j] = *(const v4f*)(&s_ab[D_C + lane * 16 + j * 4]);
        b2c[j] = *(const v4f*)(&s_ab[D_E + lane * 16 + j * 4]);
        b2t[j] = *(const v4f*)(&s_ab[D_E + D_C + lane * 16 + j * 4]);
    }

    const int stride = NBLK * WPB;                       // 8192 rows per sweep
    for (int row = blockIdx.x * WPB + wave; row < T_TOK; row += stride) {
        const int i0 = x0[row];
        const int i1 = x1[row];
        const int ip = pos_ids[row];

        // Prefetch next iteration's big-table row into the caches (random gather).
        const int nrow = row + stride;
        if (nrow < T_TOK) {
            const int ni0 = x0[nrow];
            __builtin_prefetch(emb0 + (size_t)ni0 * D_C + lane * 16, 0, 3);
        }

        const v4f* p0 = (const v4f*)(emb0      + (size_t)i0 * D_C) + lane * 4;
        const v4f* p1 = (const v4f*)(emb1      + (size_t)i1 * D_C) + lane * 4;
        const v4f* pp = (const v4f*)(pos_table + (size_t)ip * D_C) + lane * 4;

        v4f c[4], t[4];
        float sum = 0.0f, ss = 0.0f;
#pragma unroll
        for (int j = 0; j < 4; ++j) {           // b128 loads, 3 tables
            v4f e0 = p0[j];
            v4f e1 = p1[j];
            c[j] = e0 + e1;                      // content = emb0[x0] + emb1[x1]
            t[j] = pp[j];                        // timing  = pos_table[pos_ids]
        }
#pragma unroll
        for (int j = 0; j < 4; ++j) {
#pragma unroll
            for (int k = 0; k < 4; ++k) {
                const float vc = c[j][k], vt = t[j][k];
                sum += vc + vt;
                ss  += vc * vc + vt * vt;
            }
        }
        // wave32 all-reduce (5 steps)
#pragma unroll
        for (int m = 16; m >= 1; m >>= 1) {
            sum += __shfl_xor(sum, m, 32);
            ss  += __shfl_xor(ss,  m, 32);
        }
        const float mu = sum * (1.0f / D_E);
        float var = (ss - sum * mu) * (1.0f / (D_E - 1));   // unbiased (ddof=1)
        var = fmaxf(var, 0.0f);
        const float rs = 1.0f / (sqrtf(var) + EPS);          // /(sigma + eps)

        v4f* annc = (v4f*)(ann + (size_t)row * D_E) + lane * 4;
        v4f* annt = annc + (D_C / 4);                        // +512 floats
        v4f* tout = (v4f*)(timing_out + (size_t)row * D_C) + lane * 4;
#pragma unroll
        for (int j = 0; j < 4; ++j) {
            v4f oc = (c[j] - mu) * rs * a2c[j] + b2c[j];
            v4f ot = (t[j] - mu) * rs * a2t[j] + b2t[j];
            // streamed outputs: NT stores keep the 103MB emb0 table resident in L2
            __builtin_nontemporal_store(oc,  annc + j);
            __builtin_nontemporal_store(ot,  annt + j);
            __builtin_nontemporal_store(t[j], tout + j);
        }
    }
}

extern "C" void kernel_launch(void* const* d_in, const int* in_sizes, int n_in,
                              void* d_out, int out_size, void* d_ws, size_t ws_size,
                              hipStream_t stream) {
    (void)in_sizes; (void)n_in; (void)out_size; (void)d_ws; (void)ws_size;
    const int*   x0        = (const int*)d_in[0];
    const int*   x1        = (const int*)d_in[1];
    const int*   pos_ids   = (const int*)d_in[2];
    const float* emb0      = (const float*)d_in[3];
    const float* emb1      = (const float*)d_in[4];
    const float* pos_table = (const float*)d_in[5];
    const float* a2        = (const float*)d_in[6];
    const float* b2        = (const float*)d_in[7];
    float* ann    = (float*)d_out;                       // [T, 1024]
    float* timing = ann + (size_t)T_TOK * D_E;           // [T, 512]
    mle_kernel<<<NBLK, 256, 0, stream>>>(x0, x1, pos_ids, emb0, emb1, pos_table,
                                         a2, b2, ann, timing);
}